// ScaleDecoupledFAPB_36953898615190
// MI455X (gfx1250) — compile-verified
//
#include <hip/hip_runtime.h>
#include <math.h>

// ---------------------------------------------------------------------------
// ScaleDecoupledFAPB — relative positional bias MLP, MI455X (gfx1250).
//
// Roofline reasoning:
//  * Naive: 2.65M pair-head MLP evals (~12 GFLOP + 255M transcendentals).
//  * rel[n,m] depends only on (di,dj) in [-23,23]^2 -> 2209 unique inputs.
//    Kernel 1: MLP for only [8,47,47] table via V_WMMA_F32_16X16X4_F32.
//    Kernel 2: broadcast table -> [8,576,576] (10.6 MB stores). At 23.3 TB/s
//    this ~0.5us store stream is the whole workload; kernel 1 is noise, so
//    full-f32 WMMA (bit-faithful to the f32 reference) is the right choice.
// ---------------------------------------------------------------------------

#define HEADS   8
#define NFREQ   8
#define HID     64
#define IN_DIM  34
#define KPAD    36            // pad K 34 -> 36 (9 WMMA k-steps of 4)
#define GRID_H  24
#define GRID_W  24
#define NPTS    (GRID_H * GRID_W)    // 576
#define NDELTA  (2 * GRID_H - 1)     // 47
#define NPAIR   (NDELTA * NDELTA)    // 2209
#define TILES_M ((NPAIR + 15) / 16)  // 139
#define LN_EPS  1e-5f
#define PI_F    3.14159265358979323846f
#define CSTRIDE 68            // padded h1-tile row stride (bank-conflict free)

typedef float v2f __attribute__((ext_vector_type(2)));
typedef float v8f __attribute__((ext_vector_type(8)));

// ---------------------------------------------------------------------------
// Kernel 1: one wave32 per (16-row tile, head). Computes feat[16][36] and
// W1[h][36][64] into LDS, runs 4 N-tiles x 9 K-steps of
// v_wmma_f32_16x16x4_f32, then per-row GELU -> LayerNorm -> dot(W2).
// ---------------------------------------------------------------------------
__global__ __launch_bounds__(32)
void fapb_table_kernel(const float* __restrict__ freq_bands,  // [8,8]
                       const float* __restrict__ W1,          // [8,34,64]
                       const float* __restrict__ b1,          // [8,64]
                       const float* __restrict__ ln_gamma,    // [8,64]
                       const float* __restrict__ ln_beta,     // [8,64]
                       const float* __restrict__ W2,          // [8,64]
                       const float* __restrict__ b2,          // [8]
                       float* __restrict__ table)             // [8,2209]
{
    __shared__ float ldsB[KPAD][HID];      // W1[h], rows 34..35 zero
    __shared__ float ldsA[16][KPAD];       // feat tile (stride 36: conflict-free)
    __shared__ float ldsC[16][CSTRIDE];    // h1 tile   (stride 68: conflict-free)

    const int h    = blockIdx.y;
    const int tile = blockIdx.x;
    const int tid  = threadIdx.x;          // 0..31

    // ---- stage W1[h] (padded) into LDS -----------------------------------
    const float* w1g = W1 + h * IN_DIM * HID;
    for (int idx = tid; idx < KPAD * HID; idx += 32)
        (&ldsB[0][0])[idx] = (idx < IN_DIM * HID) ? w1g[idx] : 0.0f;

    // ---- compute feature tile feat[16][36] into LDS ----------------------
    for (int idx = tid; idx < 16 * KPAD; idx += 32) {
        const int r = idx / KPAD, k = idx % KPAD;
        const int p = tile * 16 + r;
        float v = 0.0f;
        if (p < NPAIR && k < IN_DIM) {
            const int a  = p / NDELTA;          // di + 23
            const int bb = p % NDELTA;          // dj + 23
            const float rel_h = (float)(a  - (GRID_H - 1)) * (2.0f / (float)(GRID_H - 1));
            const float rel_w = (float)(bb - (GRID_W - 1)) * (2.0f / (float)(GRID_W - 1));
            if (k == 0)      v = rel_h;
            else if (k == 1) v = rel_w;
            else {
                const int kk  = k - 2;
                const int grp = kk >> 3;        // 0:sin_h 1:cos_h 2:sin_w 3:cos_w
                const int f   = kk & 7;
                const float base = (grp < 2) ? rel_h : rel_w;
                const float ang  = base * freq_bands[h * NFREQ + f] * PI_F;
                v = (grp & 1) ? cosf(ang) : sinf(ang);
            }
        }
        ldsA[0][idx] = v;   // flat: idx == r*KPAD + k
    }
    __syncthreads();

    // ---- WMMA: D = A(16x36) x B(36x64), f32 x f32 -> f32 -----------------
    // ISA 7.12.2 layouts: A 16x4 f32: lanes 0-15 M=lane {K=k0,k0+1}, lanes
    // 16-31 M=lane-16 {K=k0+2,k0+3}. B 4x16: lane=N (mod 16), half-wave
    // selects K-pair. C/D 16x16: VGPR r -> M=r (+8 for upper half-wave).
    const int m16   = tid & 15;
    const int khalf = (tid < 16) ? 0 : 2;

    v2f afrag[9];
    #pragma unroll
    for (int ks = 0; ks < 9; ++ks) {
        afrag[ks].x = ldsA[m16][4 * ks + khalf + 0];
        afrag[ks].y = ldsA[m16][4 * ks + khalf + 1];
    }

    v8f acc[4];
    #pragma unroll
    for (int t = 0; t < 4; ++t) {
        v8f c = {};
        #pragma unroll
        for (int ks = 0; ks < 9; ++ks) {
            v2f bfrag;
            bfrag.x = ldsB[4 * ks + khalf + 0][16 * t + m16];
            bfrag.y = ldsB[4 * ks + khalf + 1][16 * t + m16];
            // (neg_a, A, neg_b, B, c_mod, C, reuse_a, reuse_b)
            c = __builtin_amdgcn_wmma_f32_16x16x4_f32(
                    false, afrag[ks], false, bfrag, (short)0, c, false, false);
        }
        acc[t] = c;
    }

    // ---- spill 16x64 h1 tile to LDS per C-layout -------------------------
    const int mrow = (tid < 16) ? 0 : 8;
    #pragma unroll
    for (int t = 0; t < 4; ++t)
        #pragma unroll
        for (int r = 0; r < 8; ++r)
            ldsC[mrow + r][16 * t + m16] = acc[t][r];
    __syncthreads();

    // ---- epilogue: GELU(exact) -> LayerNorm -> dot(W2) per row -----------
    if (tid < 16) {
        const int p = tile * 16 + tid;
        if (p < NPAIR) {
            const float* b1h = b1       + h * HID;
            const float* gh  = ln_gamma + h * HID;
            const float* bh  = ln_beta  + h * HID;
            const float* w2h = W2       + h * HID;
            float sum = 0.0f, sumsq = 0.0f;
            for (int d = 0; d < HID; ++d) {
                const float x = ldsC[tid][d] + b1h[d];
                const float g = 0.5f * x * (1.0f + erff(x * 0.70710678118654752f));
                ldsC[tid][d] = g;
                sum += g; sumsq += g * g;
            }
            const float mu  = sum * (1.0f / HID);
            const float var = sumsq * (1.0f / HID) - mu * mu;
            const float inv = rsqrtf(var + LN_EPS);
            float o = 0.0f;
            for (int d = 0; d < HID; ++d) {
                const float gn = (ldsC[tid][d] - mu) * inv * gh[d] + bh[d];
                o += gn * w2h[d];
            }
            table[h * NPAIR + p] = o + b2[h];
        }
    }
}

// ---------------------------------------------------------------------------
// Kernel 2: broadcast table[8,47,47] -> out[8,576,576]. Pure bandwidth:
// aligned float4 stores (GRID_W % 4 == 0 keeps the 4-span inside one dj row),
// gathers hit the 69 KB table resident in L2.
// ---------------------------------------------------------------------------
__global__ __launch_bounds__(256)
void fapb_scatter_kernel(const float* __restrict__ table, float* __restrict__ out)
{
    const int M4     = NPTS / 4;        // 144 quads per row
    const int PER_H  = NPTS * M4;       // 82944
    const int q = blockIdx.x * blockDim.x + threadIdx.x;
    if (q >= HEADS * PER_H) return;

    const int h   = q / PER_H;
    const int rem = q - h * PER_H;
    const int n   = rem / M4;
    const int m   = (rem - n * M4) * 4;

    const int i_n = n / GRID_W, j_n = n - i_n * GRID_W;
    const int i_m = m / GRID_W, j_m = m - i_m * GRID_W;
    const int a    = i_m - i_n + (GRID_H - 1);
    const int bidx = j_m - j_n + (GRID_W - 1);

    const float* src = table + h * NPAIR + a * NDELTA + bidx;
    const float4 v = make_float4(src[0], src[1], src[2], src[3]);
    float4* dst = (float4*)(out + ((size_t)(h * NPTS + n)) * (size_t)NPTS + m);
    *dst = v;
}

// ---------------------------------------------------------------------------
extern "C" void kernel_launch(void* const* d_in, const int* in_sizes, int n_in,
                              void* d_out, int out_size, void* d_ws, size_t ws_size,
                              hipStream_t stream)
{
    // setup_inputs() order: H, W, freq_bands, W1, b1, ln_gamma, ln_beta, W2, b2
    const float* freq  = (const float*)d_in[2];
    const float* W1    = (const float*)d_in[3];
    const float* b1    = (const float*)d_in[4];
    const float* gam   = (const float*)d_in[5];
    const float* bet   = (const float*)d_in[6];
    const float* W2    = (const float*)d_in[7];
    const float* b2    = (const float*)d_in[8];
    float*       out   = (float*)d_out;
    float*       table = (float*)d_ws;   // 8*2209*4 = 70,688 B of scratch

    dim3 g1(TILES_M, HEADS);
    fapb_table_kernel<<<g1, 32, 0, stream>>>(freq, W1, b1, gam, bet, W2, b2, table);

    const int total = HEADS * NPTS * (NPTS / 4);
    fapb_scatter_kernel<<<(total + 255) / 256, 256, 0, stream>>>(table, out);
}